// Encoder_76768245448827
// MI455X (gfx1250) — compile-verified
//
#include <hip/hip_runtime.h>
#include <hip/hip_bf16.h>
#include <math.h>

// ---------------------------------------------------------------------------
// Graph-transformer encoder for MI455X (gfx1250, wave32).
// Matmuls on V_WMMA_F32_16X16X4_F32 (fp32-exact); LDS staging via
// GLOBAL_LOAD_ASYNC_TO_LDS_B128 (ASYNCcnt) per CDNA5 ISA ch.10/8.
// ---------------------------------------------------------------------------

typedef __attribute__((ext_vector_type(2))) float v2f;
typedef __attribute__((ext_vector_type(8))) float v8f;

#define B 16
#define M 512
#define H 256
#define HEADS 8
#define DHEAD 32
#define ROWS (B * M)          // 8192
#define TOPK 32

__device__ __forceinline__ v8f wmma_f32(v2f a, v2f b, v8f c) {
    // D = A(16x4) * B(4x16) + C(16x16), all fp32
    return __builtin_amdgcn_wmma_f32_16x16x4_f32(false, a, false, b,
                                                 (short)0, c, false, false);
}

// Async global->LDS DMA, 16B per lane. lds_off = byte offset within the
// workgroup LDS allocation (VDST operand per CDNA5 VGLOBAL async encoding).
// The LDS pointer is passed as a (template-unused) operand so the buffer's
// address ESCAPES into the asm: with the "memory" clobber, LLVM must assume
// the asm stores through it, keeping/ordering the later ds_load reads.
__device__ __forceinline__ void async_ld_b128(unsigned lds_off, void* lds_ptr,
                                              const void* gaddr) {
    asm volatile("global_load_async_to_lds_b128 %0, %1, off"
                 :: "v"(lds_off), "v"(gaddr), "v"(lds_ptr) : "memory");
}
__device__ __forceinline__ void wait_async0() {
    asm volatile("s_wait_asynccnt 0" ::: "memory");
}

__device__ __forceinline__ float mish_f(float x) {
    float sp = log1pf(expf(x));          // overflow-safe: inf -> tanhf(inf)=1
    return x * tanhf(sp);
}

// ---------------------------------------------------------------------------
// Kernel 1: top-K edge weights (stable double-argsort rank) + row normalize.
// ---------------------------------------------------------------------------
__global__ __launch_bounds__(128) void ew_topk_kernel(
        const float* __restrict__ edge, float* __restrict__ ew) {
    __shared__ float row[M];
    __shared__ float partial[128];
    const int rid = blockIdx.x;                       // b*512 + m
    const float* e = edge + (size_t)rid * M;
    for (int n = threadIdx.x; n < M; n += 128) row[n] = e[n];
    __syncthreads();

    float w[4];
    float s = 0.f;
#pragma unroll
    for (int u = 0; u < 4; ++u) {
        const int i = threadIdx.x + u * 128;
        const float vi = row[i];
        int cnt = 0;
        for (int j = 0; j < M; ++j) {
            const float vj = row[j];
            cnt += (vj > vi) || ((vj == vi) && (j < i));
        }
        w[u] = (cnt < TOPK) ? vi : 0.f;
        s += w[u];
    }
    partial[threadIdx.x] = s;
    __syncthreads();
    for (int off = 64; off > 0; off >>= 1) {
        if (threadIdx.x < off) partial[threadIdx.x] += partial[threadIdx.x + off];
        __syncthreads();
    }
    const float inv = 1.f / (partial[0] + 1e-5f);
#pragma unroll
    for (int u = 0; u < 4; ++u) {
        const int i = threadIdx.x + u * 128;
        ew[(size_t)rid * M + i] = w[u] * inv;
    }
}

// ---------------------------------------------------------------------------
// Kernel 2: LayerNorm, torch-style (unbiased var, sqrt(var)+eps), wave/row.
// ---------------------------------------------------------------------------
__global__ __launch_bounds__(256) void ln_kernel(
        const float* __restrict__ x, float* __restrict__ out,
        const float* __restrict__ ga, const float* __restrict__ gb) {
    const int row  = blockIdx.x * 8 + (threadIdx.x >> 5);
    const int lane = threadIdx.x & 31;
    const float* xr = x + (size_t)row * H;
    float v[8];
    float s = 0.f;
#pragma unroll
    for (int u = 0; u < 8; ++u) { v[u] = xr[lane + u * 32]; s += v[u]; }
#pragma unroll
    for (int off = 16; off > 0; off >>= 1) s += __shfl_xor(s, off, 32);
    const float mean = s * (1.f / 256.f);
    float q = 0.f;
#pragma unroll
    for (int u = 0; u < 8; ++u) { const float d = v[u] - mean; q += d * d; }
#pragma unroll
    for (int off = 16; off > 0; off >>= 1) q += __shfl_xor(q, off, 32);
    const float inv = 1.f / (sqrtf(q * (1.f / 255.f)) + 1e-6f);
    float* orr = out + (size_t)row * H;
#pragma unroll
    for (int u = 0; u < 8; ++u) {
        const int c = lane + u * 32;
        orr[c] = ga[c] * (v[u] - mean) * inv + gb[c];
    }
}

// ---------------------------------------------------------------------------
// Kernel 3: GEMM  out[m,n] = act( A[m,:] . W[n,:] + bias[n] ) (+ add[m,n])
// M=8192, N=K=256.  Block = 256 thr (8 waves): 128x32 output block;
// each wave computes 16x32 (2 WMMA per A-load -> halved A traffic).
// 32x256 weight slice async-DMA'd into LDS (260-float padded rows).
// ---------------------------------------------------------------------------
#define WLD 260
__global__ __launch_bounds__(256) void gemm_kernel(
        const float* __restrict__ A, const float* __restrict__ W,
        const float* __restrict__ bias, const float* __restrict__ add,
        float* __restrict__ out, int act) {
    __shared__ float Wl[32 * WLD];                   // sole LDS object -> base 0
    const int lane = threadIdx.x & 31;
    const int wv   = threadIdx.x >> 5;
    const int n0   = blockIdx.x * 32;
    const int m0   = blockIdx.y * 128 + wv * 16;

    // async-stage W[n0..n0+31][0..255] -> LDS, 16B chunks (rows 16B-aligned:
    // WLD*4 = 1040 = 65*16)
    for (int c = threadIdx.x; c < (32 * 256) / 4; c += 256) {
        const int r = c >> 6, dd = (c & 63) << 2;
        async_ld_b128((unsigned)((r * WLD + dd) * 4), &Wl[r * WLD + dd],
                      W + (size_t)(n0 + r) * H + dd);
    }
    wait_async0();
    __syncthreads();

    const int koff = (lane >> 4) << 1;               // 0 or 2 (A/B K-pair)
    const float* Ap = A + (size_t)(m0 + (lane & 15)) * H + koff;
    const float* B0 = &Wl[(lane & 15) * WLD + koff];
    const float* B1 = &Wl[((lane & 15) + 16) * WLD + koff];

    v8f acc0 = {}, acc1 = {};
#pragma unroll 8
    for (int kt = 0; kt < 64; ++kt) {
        const v2f a  = *(const v2f*)(Ap + kt * 4);
        const v2f b0 = *(const v2f*)(B0 + kt * 4);
        const v2f b1 = *(const v2f*)(B1 + kt * 4);
        acc0 = wmma_f32(a, b0, acc0);
        acc1 = wmma_f32(a, b1, acc1);
    }

    const int c0 = n0 + (lane & 15);
    const int rb = m0 + ((lane >> 4) << 3);          // +8 for upper half-wave
    const float bs0 = bias[c0];
    const float bs1 = bias[c0 + 16];
#pragma unroll
    for (int r = 0; r < 8; ++r) {
        float v0 = acc0[r] + bs0;
        float v1 = acc1[r] + bs1;
        if (act) { v0 = mish_f(v0); v1 = mish_f(v1); }
        const size_t o = (size_t)(rb + r) * H + c0;
        if (add) { v0 += add[o]; v1 += add[o + 16]; }
        out[o]      = v0;
        out[o + 16] = v1;
    }
}

// ---------------------------------------------------------------------------
// Kernel 4: fused attention for one (batch, head, 64-row block).
// Block = 128 thr (4 waves, 16 rows each).
// LDS: S 64x516, K 512x36, V 512x36 => 279,552 B dynamic (<= 320KB/WGP).
// K/V staged via async b128 DMA (rows 36*4=144B, 16B-aligned chunks).
// softmax( where(mask,-1e12, QK^T) * ew * (1/sqrt(d)) ), then P@V.
// ---------------------------------------------------------------------------
#define SROW 516
#define KROW 36
#define ATTN_SMEM_FLOATS (64 * SROW + 2 * M * KROW)
#define ATTN_SMEM_BYTES  (ATTN_SMEM_FLOATS * 4)

__global__ __launch_bounds__(128) void attn_kernel(
        const float* __restrict__ q, const float* __restrict__ kmat,
        const float* __restrict__ vmat, const float* __restrict__ ew,
        const int* __restrict__ masks, float* __restrict__ outb) {
    extern __shared__ float smem[];                  // dynamic LDS base = 0
    float* S  = smem;                                // 64 * 516
    float* Kl = smem + 64 * SROW;                    // 512 * 36
    float* Vl = Kl + M * KROW;                       // 512 * 36
    const unsigned kl_base = (unsigned)(64 * SROW * 4);
    const unsigned vl_base = kl_base + (unsigned)(M * KROW * 4);

    const int tid  = threadIdx.x;
    const int lane = tid & 31;
    const int wv   = tid >> 5;
    const int blk  = blockIdx.x & 7;                 // 64-row block in sequence
    const int h    = (blockIdx.x >> 3) & 7;
    const int bb   = blockIdx.x >> 6;
    const size_t base = (size_t)bb * M * H + (size_t)h * DHEAD;

    // Async-stage K,V head slices (512 x 32) into LDS, 16B chunks.
    for (int c = tid; c < (M * DHEAD) / 4; c += 128) {
        const int n = c >> 3, dd = (c & 7) << 2;
        const unsigned lo = (unsigned)((n * KROW + dd) * 4);
        const size_t   go = base + (size_t)n * H + dd;
        async_ld_b128(kl_base + lo, &Kl[n * KROW + dd], kmat + go);
        async_ld_b128(vl_base + lo, &Vl[n * KROW + dd], vmat + go);
    }

    // Q fragments for this wave's 16 rows (A-matrix layout, 8 K-steps of 4).
    const int koff = (lane >> 4) << 1;
    const int mloc = blk * 64 + wv * 16 + (lane & 15);
    const float* qp = q + base + (size_t)mloc * H + koff;
    v2f qf[8];
#pragma unroll
    for (int kk = 0; kk < 8; ++kk) qf[kk] = *(const v2f*)(qp + kk * 4);
    wait_async0();
    __syncthreads();

    // ---- scores: S[16 x 512] per wave via WMMA -----------------------------
    for (int nt = 0; nt < 32; ++nt) {
        v8f acc = {};
        const float* kp = &Kl[(nt * 16 + (lane & 15)) * KROW + koff];
#pragma unroll
        for (int kk = 0; kk < 8; ++kk) {
            const v2f bf = *(const v2f*)(kp + kk * 4);
            acc = wmma_f32(qf[kk], bf, acc);
        }
        float* sp = &S[(wv * 16 + ((lane >> 4) << 3)) * SROW + nt * 16 + (lane & 15)];
#pragma unroll
        for (int r = 0; r < 8; ++r) sp[r * SROW] = acc[r];
    }
    __syncthreads();

    // ---- softmax of where(mask,-1e12,s)*ew*scale; 2 lanes per row ----------
    {
        const int rl = tid >> 1;                     // local row 0..63
        const int half = tid & 1;
        const int mrow = blk * 64 + rl;
        const float* ewr = ew + ((size_t)bb * M + mrow) * M;
        const int* mr = masks + (size_t)bb * M;
        float* sr = S + rl * SROW;
        const float scale = 0.17677669529663687f;    // 1/sqrt(32)

        float mx = -3.4e38f;
        for (int n = half; n < M; n += 2) {
            float sv = sr[n];
            if (mr[n] == 0) sv = -1e12f;
            const float lg = sv * ewr[n] * scale;
            sr[n] = lg;
            mx = fmaxf(mx, lg);
        }
        mx = fmaxf(mx, __shfl_xor(mx, 1, 32));
        float sum = 0.f;
        for (int n = half; n < M; n += 2) {
            const float p = expf(sr[n] - mx);
            sr[n] = p;
            sum += p;
        }
        sum += __shfl_xor(sum, 1, 32);
        const float inv = 1.f / sum;
        for (int n = half; n < M; n += 2) sr[n] *= inv;
    }
    __syncthreads();

    // ---- P @ V : 16x32 per wave via WMMA (two 16-wide d tiles) -------------
    v8f o0 = {}, o1 = {};
    {
        const float* ap = &S[(wv * 16 + (lane & 15)) * SROW + koff];
        const int nb = lane & 15;
#pragma unroll 4
        for (int kt = 0; kt < 128; ++kt) {
            const int k = kt * 4;
            const v2f a = *(const v2f*)(ap + k);
            v2f b0, b1;
            b0.x = Vl[(k + koff + 0) * KROW + nb];
            b0.y = Vl[(k + koff + 1) * KROW + nb];
            b1.x = Vl[(k + koff + 0) * KROW + 16 + nb];
            b1.y = Vl[(k + koff + 1) * KROW + 16 + nb];
            o0 = wmma_f32(a, b0, o0);
            o1 = wmma_f32(a, b1, o1);
        }
    }
    const size_t orow0 = (size_t)(bb * M + blk * 64 + wv * 16 + ((lane >> 4) << 3));
#pragma unroll
    for (int r = 0; r < 8; ++r) {
        const size_t o = (orow0 + r) * H + (size_t)h * DHEAD + (lane & 15);
        outb[o]      = o0[r];
        outb[o + 16] = o1[r];
    }
}

// ---------------------------------------------------------------------------
// Host driver.  Inputs (setup_inputs order):
// 0 node_features[16,512,256] f32   1 edge_features[16,512,512] f32
// 2 masks[16,512] i32               3 attn_W[4,4,256,256] f32
// 4 attn_b[4,4,256] f32             5 ffn_W[4,2,256,256] f32
// 6 ffn_b[4,2,256] f32              7 ln_a[256] f32   8 ln_b[256] f32
// ---------------------------------------------------------------------------
extern "C" void kernel_launch(void* const* d_in, const int* in_sizes, int n_in,
                              void* d_out, int out_size, void* d_ws, size_t ws_size,
                              hipStream_t stream) {
    const float* node  = (const float*)d_in[0];
    const float* edge  = (const float*)d_in[1];
    const int*   masks = (const int*)d_in[2];
    const float* attnW = (const float*)d_in[3];
    const float* attnB = (const float*)d_in[4];
    const float* ffnW  = (const float*)d_in[5];
    const float* ffnB  = (const float*)d_in[6];
    const float* lna   = (const float*)d_in[7];
    const float* lnb   = (const float*)d_in[8];

    float* ws   = (float*)d_ws;
    float* ewb  = ws;                               // 16*512*512
    float* xn   = ewb + (size_t)B * M * M;          // LN out = residual base
    float* qb   = xn  + (size_t)ROWS * H;
    float* kb   = qb  + (size_t)ROWS * H;
    float* vb   = kb  + (size_t)ROWS * H;
    float* atb  = vb  + (size_t)ROWS * H;           // attention context
    float* xb   = atb + (size_t)ROWS * H;           // residual stream
    float* yb   = xb  + (size_t)ROWS * H;           // FFN hidden

    hipFuncSetAttribute(reinterpret_cast<const void*>(attn_kernel),
                        hipFuncAttributeMaxDynamicSharedMemorySize, ATTN_SMEM_BYTES);

    ew_topk_kernel<<<ROWS, 128, 0, stream>>>(edge, ewb);

    const dim3 ggrid(8, 64);   // N/32 x M/128
    const float* x = node;
    for (int i = 0; i < 4; ++i) {
        // x = LN(x)  (xn becomes the residual base for this layer)
        ln_kernel<<<ROWS / 8, 256, 0, stream>>>(x, xn, lna, lnb);
        const float* Wq = attnW + (size_t)(i * 4 + 0) * H * H;
        const float* Wk = attnW + (size_t)(i * 4 + 1) * H * H;
        const float* Wv = attnW + (size_t)(i * 4 + 2) * H * H;
        const float* Wo = attnW + (size_t)(i * 4 + 3) * H * H;
        const float* bq = attnB + (size_t)(i * 4 + 0) * H;
        const float* bk = attnB + (size_t)(i * 4 + 1) * H;
        const float* bv = attnB + (size_t)(i * 4 + 2) * H;
        const float* bo = attnB + (size_t)(i * 4 + 3) * H;
        gemm_kernel<<<ggrid, 256, 0, stream>>>(xn, Wq, bq, nullptr, qb, 0);
        gemm_kernel<<<ggrid, 256, 0, stream>>>(xn, Wk, bk, nullptr, kb, 0);
        gemm_kernel<<<ggrid, 256, 0, stream>>>(xn, Wv, bv, nullptr, vb, 0);
        attn_kernel<<<B * HEADS * (M / 64), 128, ATTN_SMEM_BYTES, stream>>>(
            qb, kb, vb, ewb, masks, atb);
        // x = xn + mish(atb @ Wo + bo)
        gemm_kernel<<<ggrid, 256, 0, stream>>>(atb, Wo, bo, xn, xb, 1);
        // x = LN(x)
        ln_kernel<<<ROWS / 8, 256, 0, stream>>>(xb, xn, lna, lnb);
        // y = mish(xn @ Wf0 + b0);  x = xn + mish(y @ Wf1 + b1)
        const float* Wf0 = ffnW + (size_t)(i * 2 + 0) * H * H;
        const float* Wf1 = ffnW + (size_t)(i * 2 + 1) * H * H;
        const float* bf0 = ffnB + (size_t)(i * 2 + 0) * H;
        const float* bf1 = ffnB + (size_t)(i * 2 + 1) * H;
        gemm_kernel<<<ggrid, 256, 0, stream>>>(xn, Wf0, bf0, nullptr, yb, 1);
        float* outp = (i == 3) ? (float*)d_out : xb;
        gemm_kernel<<<ggrid, 256, 0, stream>>>(yb, Wf1, bf1, xn, outp, 1);
        x = xb;
    }
}